// messagePassingBondConv_5995774345719
// MI455X (gfx1250) — compile-verified
//
#include <hip/hip_runtime.h>

typedef __attribute__((ext_vector_type(2))) float v2f;
typedef __attribute__((ext_vector_type(4))) float v4f;
typedef __attribute__((ext_vector_type(8))) float v8f;

#define F 128
#define LDSROW 132              // padded row stride (floats): 132%64=4 -> conflict-free b64 frag loads
#define TILEF (16 * LDSROW)    // one 16x128 tile (padded)

__device__ __forceinline__ v8f v8zero() {
  v8f v;
#pragma unroll
  for (int i = 0; i < 8; ++i) v[i] = 0.0f;
  return v;
}

__device__ __forceinline__ float fast_sigmoid(float t) {
  return 1.0f / (1.0f + __expf(-t));
}
__device__ __forceinline__ float fast_tanh(float t) {
  return 1.0f - 2.0f / (__expf(2.0f * t) + 1.0f);
}

// D = A(16x4 f32) * B(4x16 f32) + C(16x16 f32)
__device__ __forceinline__ v8f wmma_f32(v2f a, v2f b, v8f c) {
  return __builtin_amdgcn_wmma_f32_16x16x4_f32(
      /*neg_a=*/false, a, /*neg_b=*/false, b,
      /*c_mod=*/(short)0, c, /*reuse_a=*/false, /*reuse_b=*/false);
}

__global__ __launch_bounds__(128) void bondconv_gru_kernel(
    const float* __restrict__ x,          // (N,128)
    const float* __restrict__ bond_x,     // (E,128)
    const int*   __restrict__ pairsNext,  // (E,2)
    const int*   __restrict__ pairsPrev,  // (E,2)
    const float* __restrict__ wNext,      // (128,128)
    const float* __restrict__ wPrev,      // (128,128)
    const float* __restrict__ bNext,      // (1,128)
    const float* __restrict__ bPrev,      // (1,128)
    const float* __restrict__ bn_gamma,   // (128,)
    const float* __restrict__ bn_beta,    // (128,)
    const float* __restrict__ bn_mean,    // (128,)
    const float* __restrict__ bn_var,     // (128,)
    const float* __restrict__ gk,         // (128,384)
    const float* __restrict__ grk,        // (128,384)
    const float* __restrict__ gbias,      // (2,384)
    float* __restrict__ out,              // (E,128)
    int E)
{
  extern __shared__ v4f smem4[];
  float* smem = (float*)smem4;

  const int lane = threadIdx.x & 31;
  const int wave = threadIdx.x >> 5;
  const int half = lane >> 4;   // 0: lanes 0-15, 1: lanes 16-31
  const int nl   = lane & 15;

  float* BOND = smem + wave * (2 * TILEF);  // bond tile (persists)
  float* S    = BOND + TILEF;               // gather+bond, later aggre

  const int tile = blockIdx.x * 4 + wave;
  const long e0 = (long)tile * 16;
  if (e0 >= E) return;  // uniform per wave -> EXEC stays all-ones for WMMA

  // ---- stage bond tile into LDS (coalesced: 32 lanes x 16B per row) ----
  for (int r = 0; r < 16; ++r) {
    long e = e0 + r; if (e >= E) e = E - 1;
    v4f bv = *(const v4f*)(bond_x + e * F + lane * 4);
    *(v4f*)(BOND + r * LDSROW + lane * 4) = bv;
  }

  // ---- phase 1: acc = (next_x+bond)@wNext + (prev_x+bond)@wPrev ----
  v8f acc[8];
#pragma unroll
  for (int nt = 0; nt < 8; ++nt) acc[nt] = v8zero();

  for (int pass = 0; pass < 2; ++pass) {
    const float* W     = pass ? wPrev     : wNext;
    const int*   pairs = pass ? pairsPrev : pairsNext;

    // S = x[gather] + bond  (gathered rows are coalesced 512B segments)
    for (int r = 0; r < 16; ++r) {
      long e = e0 + r; if (e >= E) e = E - 1;
      int src = pairs[e * 2 + 1];
      v4f xv = *(const v4f*)(x + (long)src * F + lane * 4);
      v4f bv = *(const v4f*)(BOND + r * LDSROW + lane * 4);
      *(v4f*)(S + r * LDSROW + lane * 4) = xv + bv;
    }

    for (int kb = 0; kb < 32; ++kb) {
      const int k0 = kb * 4 + half * 2;
      // A frag: lane l -> (M=l%16, K=k0..k0+1)
      v2f a = *(const v2f*)(S + nl * LDSROW + k0);
      const float* w0 = W + (long)k0 * F + nl;
#pragma unroll
      for (int nt = 0; nt < 8; ++nt) {
        v2f b;                      // B frag: V0 = rows k0/k0+2, V1 = k0+1/k0+3
        b.x = w0[nt * 16];
        b.y = w0[F + nt * 16];
        acc[nt] = wmma_f32(a, b, acc[nt]);
      }
    }
  }

  // ---- epilogue 1: +bias, ReLU, BatchNorm -> AGG (overwrites S) ----
#pragma unroll
  for (int nt = 0; nt < 8; ++nt) {
    const int col = nt * 16 + nl;
    float bsum  = bNext[col] + bPrev[col];
    float scale = bn_gamma[col] * rsqrtf(bn_var[col] + 1e-3f);
    float shift = bn_beta[col] - bn_mean[col] * scale;
#pragma unroll
    for (int i = 0; i < 8; ++i) {   // C layout: VGPR i -> row i (+8 for hi half)
      float v = acc[nt][i] + bsum;
      v = v > 0.0f ? v : 0.0f;
      S[(i + half * 8) * LDSROW + col] = v * scale + shift;
    }
  }

  // ---- phase 2: GRU  mx = AGG@gk + b0, mh = BOND@grk + b1 ----
  for (int nt = 0; nt < 8; ++nt) {
    v8f xz = v8zero(), xr = v8zero(), xh = v8zero();
    v8f hz = v8zero(), hr = v8zero(), hh = v8zero();
    const int colb = nt * 16 + nl;

    for (int kb = 0; kb < 32; ++kb) {
      const int k0 = kb * 4 + half * 2;
      v2f aA = *(const v2f*)(S    + nl * LDSROW + k0);   // aggre frag
      v2f aH = *(const v2f*)(BOND + nl * LDSROW + k0);   // bond  frag
      const float* kp = gk  + (long)k0 * 384 + colb;
      const float* rp = grk + (long)k0 * 384 + colb;
      v2f b;
      b.x = kp[0];        b.y = kp[384];        xz = wmma_f32(aA, b, xz);
      b.x = kp[128];      b.y = kp[384 + 128];  xr = wmma_f32(aA, b, xr);
      b.x = kp[256];      b.y = kp[384 + 256];  xh = wmma_f32(aA, b, xh);
      b.x = rp[0];        b.y = rp[384];        hz = wmma_f32(aH, b, hz);
      b.x = rp[128];      b.y = rp[384 + 128];  hr = wmma_f32(aH, b, hr);
      b.x = rp[256];      b.y = rp[384 + 256];  hh = wmma_f32(aH, b, hh);
    }

    const float biz = gbias[colb];
    const float bir = gbias[128 + colb];
    const float bih = gbias[256 + colb];
    const float brz = gbias[384 + colb];
    const float brr = gbias[384 + 128 + colb];
    const float brh = gbias[384 + 256 + colb];

#pragma unroll
    for (int i = 0; i < 8; ++i) {
      const int row = i + half * 8;
      const long e = e0 + row;
      float h  = BOND[row * LDSROW + colb];
      float z  = fast_sigmoid(xz[i] + biz + hz[i] + brz);
      float r  = fast_sigmoid(xr[i] + bir + hr[i] + brr);
      float hc = fast_tanh(xh[i] + bih + r * (hh[i] + brh));
      float o  = z * h + (1.0f - z) * hc;
      if (e < E) out[e * F + colb] = o;
    }
  }
}

extern "C" void kernel_launch(void* const* d_in, const int* in_sizes, int n_in,
                              void* d_out, int out_size, void* d_ws, size_t ws_size,
                              hipStream_t stream) {
  const float* x         = (const float*)d_in[0];
  const float* bond_x    = (const float*)d_in[1];
  const int*   pairsNext = (const int*)  d_in[2];
  const int*   pairsPrev = (const int*)  d_in[3];
  const float* wNext     = (const float*)d_in[4];
  const float* wPrev     = (const float*)d_in[5];
  const float* bNext     = (const float*)d_in[6];
  const float* bPrev     = (const float*)d_in[7];
  const float* bn_gamma  = (const float*)d_in[8];
  const float* bn_beta   = (const float*)d_in[9];
  const float* bn_mean   = (const float*)d_in[10];
  const float* bn_var    = (const float*)d_in[11];
  const float* gk        = (const float*)d_in[12];
  const float* grk       = (const float*)d_in[13];
  const float* gbias     = (const float*)d_in[14];

  const int E = in_sizes[1] / F;                 // bond_x is (E,128)
  const int tiles  = (E + 15) / 16;
  const int blocks = (tiles + 3) / 4;            // 4 waves (tiles) per block
  const size_t shmem = (size_t)4 * 2 * TILEF * sizeof(float);  // 67,584 B

  bondconv_gru_kernel<<<blocks, 128, shmem, stream>>>(
      x, bond_x, pairsNext, pairsPrev, wNext, wPrev, bNext, bPrev,
      bn_gamma, bn_beta, bn_mean, bn_var, gk, grk, gbias,
      (float*)d_out, E);
}